// VectorQuantizer_74259984547867
// MI455X (gfx1250) — compile-verified
//
#include <hip/hip_runtime.h>
#include <math.h>

typedef float v2f __attribute__((ext_vector_type(2)));
typedef float v8f __attribute__((ext_vector_type(8)));

#define VQ_T 4096
#define VQ_D 64
#define VQ_K 512
#define VQ_B 16
#define VQ_N (VQ_B * VQ_T)       // 65536 tokens
#define VQ_PAIRS (VQ_N / 32)     // 2048 pairs of 16-token tiles
#define QELEMS (VQ_N * VQ_D)     // 4194304 quantized elements

// ---------------------------------------------------------------------------
// Kernel 1: zero the histogram, precompute 0.5*||e_k||^2 per code.
// ---------------------------------------------------------------------------
__global__ void vq_init(const float* __restrict__ cb,
                        unsigned* __restrict__ counts,
                        float* __restrict__ hn) {
    int t = blockIdx.x * blockDim.x + threadIdx.x;
    if (t < VQ_K) {
        counts[t] = 0u;
        const float* row = cb + t * VQ_D;
        float s = 0.f;
#pragma unroll
        for (int d = 0; d < VQ_D; ++d) s += row[d] * row[d];
        hn[t] = 0.5f * s;
    }
}

// ---------------------------------------------------------------------------
// Kernel 2: WMMA nearest-code search + fused outputs.
// One wave = TWO tiles of 16 tokens: one set of B (codebook) loads feeds two
// independent V_WMMA_F32_16X16X4_F32 accumulator chains (ILP + 2x B reuse).
// ---------------------------------------------------------------------------
__global__ __launch_bounds__(256) void vq_main(
    const float* __restrict__ in, const float* __restrict__ cb,
    const float* __restrict__ hn, unsigned* __restrict__ counts,
    float* __restrict__ partials, float* __restrict__ out_q,
    float* __restrict__ out_enc) {
    const int lane   = threadIdx.x & 31;
    const int wave   = (blockIdx.x * blockDim.x + threadIdx.x) >> 5;
    const int nwaves = (gridDim.x * blockDim.x) >> 5;
    const int token  = lane & 15;   // M (and N) index within a tile
    const int h      = lane >> 4;   // K-half selector per WMMA f32 layout

    float acc = 0.f;  // sum of (q - x)^2 over this wave's elements

    for (int pair = wave; pair < VQ_PAIRS; pair += nwaves) {
        const int n0 = pair * 32;        // first token of tile 0
        const int bb = n0 >> 12;         // batch index (32 divides 4096)
        const int t0 = n0 & 4095;

        // ---- Load A for both tiles: a[p][i] = x[tok, 4i+2h], x[tok,4i+2h+1]
        v2f a[2][16];
#pragma unroll
        for (int p = 0; p < 2; ++p) {
            const float* __restrict__ xin =
                in + (size_t)bb * VQ_D * VQ_T + t0 + 16 * p + token;
#pragma unroll
            for (int i = 0; i < 16; ++i) {
                const int d0 = 4 * i + 2 * h;
                a[p][i].x = xin[(size_t)d0 * VQ_T];
                a[p][i].y = xin[(size_t)(d0 + 1) * VQ_T];
            }
        }

        float best[2][8];
        int   bidx[2][8];
#pragma unroll
        for (int p = 0; p < 2; ++p)
#pragma unroll
            for (int r = 0; r < 8; ++r) {
                best[p][r] = -3.0e38f;
                bidx[p][r] = 0;
            }

        // ---- 32 code sub-tiles of 16 codes each ----
        for (int ct = 0; ct < 32; ++ct) {
            const int code = ct * 16 + token;  // this lane's column (code)
            const float* __restrict__ crow = cb + code * VQ_D + 2 * h;
            v2f bm[16];
#pragma unroll
            for (int i = 0; i < 16; ++i)
                bm[i] = *(const v2f*)(crow + 4 * i);  // aligned b64 loads

            v8f c0 = {0.f, 0.f, 0.f, 0.f, 0.f, 0.f, 0.f, 0.f};
            v8f c1 = {0.f, 0.f, 0.f, 0.f, 0.f, 0.f, 0.f, 0.f};
#pragma unroll
            for (int i = 0; i < 16; ++i) {  // two independent WMMA chains
                c0 = __builtin_amdgcn_wmma_f32_16x16x4_f32(
                        false, a[0][i], false, bm[i], (short)0, c0, false,
                        false);
                c1 = __builtin_amdgcn_wmma_f32_16x16x4_f32(
                        false, a[1][i], false, bm[i], (short)0, c1, false,
                        false);
            }

            const float hnv = hn[code];
#pragma unroll
            for (int r = 0; r < 8; ++r) {
                const float s0 = c0[r] - hnv;  // row M = r + 8*h, col = code
                if (s0 > best[0][r]) { best[0][r] = s0; bidx[0][r] = code; }
                const float s1 = c1[r] - hnv;
                if (s1 > best[1][r]) { best[1][r] = s1; bidx[1][r] = code; }
            }
        }

        // ---- argmax across the 16 code lanes (butterfly; ties -> min idx) --
#pragma unroll
        for (int m = 8; m >= 1; m >>= 1) {
#pragma unroll
            for (int p = 0; p < 2; ++p)
#pragma unroll
                for (int r = 0; r < 8; ++r) {
                    const float os = __shfl_xor(best[p][r], m, 32);
                    const int   oi = __shfl_xor(bidx[p][r], m, 32);
                    if (os > best[p][r] ||
                        (os == best[p][r] && oi < bidx[p][r])) {
                        best[p][r] = os;
                        bidx[p][r] = oi;
                    }
                }
        }

        // ---- per-tile epilogue: broadcast, histogram, one-hot, gather ----
#pragma unroll
        for (int p = 0; p < 2; ++p) {
            const int nt0 = n0 + 16 * p;

            // rows 0-7 live in lanes 0-15 (h=0), rows 8-15 in lanes 16-31
            int idx_tok[16];
#pragma unroll
            for (int r = 0; r < 8; ++r) {
                idx_tok[r]     = __shfl(bidx[p][r], 0, 32);
                idx_tok[r + 8] = __shfl(bidx[p][r], 16, 32);
            }

            // histogram (integer atomics: deterministic final counts)
            if (lane < 16) atomicAdd(&counts[idx_tok[lane]], 1u);

            // one-hot encodings: 16 rows x 512 f32, NT b64 stores (128MB
            // write-once stream -> keep it out of L2)
#pragma unroll
            for (int m = 0; m < 16; ++m) {
                float* rowp = out_enc + (size_t)(nt0 + m) * VQ_K;
                const int tgt = idx_tok[m];
#pragma unroll
                for (int j = 0; j < 8; ++j) {
                    const int c0i = j * 64 + lane * 2;
                    v2f v;
                    v.x = (c0i == tgt) ? 1.f : 0.f;
                    v.y = (c0i + 1 == tgt) ? 1.f : 0.f;
                    __builtin_nontemporal_store(v, (v2f*)(rowp + c0i));
                }
            }

            // quantized gather + write back in [B, D, T], loss accumulate
            const float* __restrict__ qrow =
                cb + idx_tok[token] * VQ_D + 2 * h;
            float* __restrict__ qout =
                out_q + (size_t)bb * VQ_D * VQ_T + (t0 + 16 * p) + token;
#pragma unroll
            for (int i = 0; i < 16; ++i) {
                const int d0 = 4 * i + 2 * h;
                const v2f q = *(const v2f*)(qrow + 4 * i);
                qout[(size_t)d0 * VQ_T]       = q.x;
                qout[(size_t)(d0 + 1) * VQ_T] = q.y;
                const float e0 = q.x - a[p][i].x;
                const float e1 = q.y - a[p][i].y;
                acc += e0 * e0 + e1 * e1;
            }
        }
    }

    // ---- per-wave deterministic partial for the loss ----
#pragma unroll
    for (int m = 16; m >= 1; m >>= 1) acc += __shfl_xor(acc, m, 32);
    if (lane == 0) partials[wave] = acc;
}

// ---------------------------------------------------------------------------
// Kernel 3: reduce partials -> loss; counts -> perplexity.
// ---------------------------------------------------------------------------
__global__ __launch_bounds__(256) void vq_final(
    const unsigned* __restrict__ counts, const float* __restrict__ partials,
    float* __restrict__ out) {
    __shared__ float red[256];
    const int tid = threadIdx.x;

    float s = 0.f;
    for (int i = tid; i < VQ_PAIRS; i += 256) s += partials[i];
    red[tid] = s;
    __syncthreads();
    for (int off = 128; off >= 1; off >>= 1) {
        if (tid < off) red[tid] += red[tid + off];
        __syncthreads();
    }
    const float losssum = red[0];
    __syncthreads();

    float ent = 0.f;
    for (int i = tid; i < VQ_K; i += 256) {
        const float p = (float)counts[i] * (1.0f / (float)VQ_N);
        ent += p * logf(p + 1e-10f);
    }
    red[tid] = ent;
    __syncthreads();
    for (int off = 128; off >= 1; off >>= 1) {
        if (tid < off) red[tid] += red[tid + off];
        __syncthreads();
    }

    if (tid == 0) {
        // loss = q_latent + 0.25 * e_latent = 1.25 * mean((q - x)^2)
        out[0] = 1.25f * losssum / (float)QELEMS;
        out[1 + QELEMS] = expf(-red[0]);  // perplexity
    }
}

// ---------------------------------------------------------------------------
// Launch. Workspace layout (floats): [0..511] counts(u32), [512..1023] hn,
// [1024..3071] per-wave partials  (12 KB total).
// Output layout: [loss(1) | quantized(4194304) | perplexity(1) | enc(33554432)]
// ---------------------------------------------------------------------------
extern "C" void kernel_launch(void* const* d_in, const int* in_sizes, int n_in,
                              void* d_out, int out_size, void* d_ws,
                              size_t ws_size, hipStream_t stream) {
    const float* in = (const float*)d_in[0];   // [16, 64, 4096] f32
    const float* cb = (const float*)d_in[1];   // [512, 64] f32
    float* out = (float*)d_out;

    unsigned* counts   = (unsigned*)d_ws;
    float*    hn       = (float*)d_ws + VQ_K;
    float*    partials = (float*)d_ws + 2 * VQ_K;

    float* out_q   = out + 1;
    float* out_enc = out + 2 + QELEMS;

    vq_init<<<2, 256, 0, stream>>>(cb, counts, hn);
    vq_main<<<256, 256, 0, stream>>>(in, cb, hn, counts, partials, out_q,
                                     out_enc);
    vq_final<<<1, 256, 0, stream>>>(counts, partials, out);
}